// ODFlowDecoder_19189913878655
// MI455X (gfx1250) — compile-verified
//
#include <hip/hip_runtime.h>
#include <hip/hip_bf16.h>

typedef __attribute__((ext_vector_type(2))) float v2f;
typedef __attribute__((ext_vector_type(8))) float v8f;

// D = A(16x4) * B(4x16) + C, fp32 WMMA (exact fp32 path on CDNA5)
#define WMMA_F32_K4(a, b, c) \
  __builtin_amdgcn_wmma_f32_16x16x4_f32(false, (a), false, (b), (short)0, (c), false, false)

// ---------------------------------------------------------------------------
// Tiny helper: out[j] = sum_k v[k] * W[k*128 + j]   (fold b2o through Wb)
// ---------------------------------------------------------------------------
__global__ void vecmat128_kernel(const float* __restrict__ v,
                                 const float* __restrict__ W,
                                 float* __restrict__ out) {
  const int j = threadIdx.x;  // 128 threads
  float acc = 0.f;
  for (int k = 0; k < 128; ++k) acc = fmaf(v[k], W[k * 128 + j], acc);
  out[j] = acc;
}

// ---------------------------------------------------------------------------
// Generic fp32 WMMA GEMM: C[M,N] = act(A[M,K] @ W[K,N] + bias[N])
// A row-major, W row-major. One 16x16 tile per wave. M%16==0, N%16==0, K%4==0.
//
// fp32 16x16x4 fragment layout (ISA 7.12.2):
//   A: lane L holds A[row0 + L%16][k + 2*(L>>4) + {0,1}]  -> float2
//   B: lane L holds W[k + 2*(L>>4) + {0,1}][col0 + L%16]  -> two strided loads
//   C: VGPR r, lane L -> C[row0 + r + 8*(L>>4)][col0 + L%16]
// ---------------------------------------------------------------------------
__global__ void gemm_wmma_kernel(const float* __restrict__ A,
                                 const float* __restrict__ W,
                                 const float* __restrict__ bias,
                                 float* __restrict__ C,
                                 int M, int N, int K, int relu) {
  const int lane = threadIdx.x & 31;
  const int wave = blockIdx.x * (blockDim.x >> 5) + (threadIdx.x >> 5);
  const int tilesN = N >> 4;
  const int tiles  = (M >> 4) * tilesN;
  if (wave >= tiles) return;  // whole-wave exit: EXEC stays all-ones for WMMA

  const int ti   = wave / tilesN;
  const int tj   = wave - ti * tilesN;
  const int row0 = ti << 4, col0 = tj << 4;
  const int l16  = lane & 15;
  const int kk   = (lane >> 4) << 1;

  const float* Ap = A + (size_t)(row0 + l16) * K + kk;  // A[row][k+kk]
  const float* Wp = W + (size_t)kk * N + (col0 + l16);  // W[k+kk][col]

  v8f c = {};
  for (int k = 0; k < K; k += 4) {
    v2f a;
    a.x = Ap[k];
    a.y = Ap[k + 1];
    v2f b;
    b.x = Wp[(size_t)k * N];
    b.y = Wp[(size_t)(k + 1) * N];
    c = WMMA_F32_K4(a, b, c);
  }

  const int   col   = col0 + l16;
  const float bv    = bias ? bias[col] : 0.f;
  const int   rbase = row0 + ((lane >> 4) << 3);
#pragma unroll
  for (int r = 0; r < 8; ++r) {
    float v = c[r] + bv;
    if (relu) v = fmaxf(v, 0.f);
    C[(size_t)(rbase + r) * N + col] = v;
  }
}

// ---------------------------------------------------------------------------
// Fused scores + horizon expansion (the bandwidth-dominant kernel):
//   scores(b,i,j) = dot(oW[b,i,:], d[b,j,:]) + bb          (K = 128)
//   out[b,t,i,j]  = relu(scores * wh[t] + bh[t])           (T = 12)
// One 16x16 score tile per wave; 32 v_wmma_f32_16x16x4_f32 per tile; the
// 12-plane epilogue streams ~201 MB straight to HBM (the roofline limiter).
// oW and d (1 MB/batch) stay resident in the 192 MB L2 across tile reuse.
// ---------------------------------------------------------------------------
__global__ void scores_expand_kernel(const float* __restrict__ oW,
                                     const float* __restrict__ dM,
                                     const float* __restrict__ wh,
                                     const float* __restrict__ bh,
                                     const float* __restrict__ bbp,
                                     float* __restrict__ out) {
  constexpr int N = 1024, T = 12, K = 128, NT = N / 16;
  const int lane = threadIdx.x & 31;
  const int wave = blockIdx.x * (blockDim.x >> 5) + (threadIdx.x >> 5);
  constexpr int tilesPerBatch = NT * NT;

  const int b   = wave / tilesPerBatch;
  const int tid = wave - b * tilesPerBatch;
  const int ti  = tid / NT;
  const int tj  = tid - ti * NT;
  const int l16 = lane & 15;
  const int kk  = (lane >> 4) << 1;

  // A = oW rows, B = d rows (A @ B^T): both fragments are contiguous float2
  const float* Ap = oW + ((size_t)b * N + ti * 16 + l16) * K + kk;
  const float* Bp = dM + ((size_t)b * N + tj * 16 + l16) * K + kk;

  v8f c = {};
#pragma unroll
  for (int k = 0; k < K; k += 4) {
    v2f a  = *(const v2f*)(Ap + k);
    v2f bf = *(const v2f*)(Bp + k);
    c = WMMA_F32_K4(a, bf, c);
  }

  const float bb = *bbp;
  float whv[T], bhv[T];
#pragma unroll
  for (int t = 0; t < T; ++t) { whv[t] = wh[t]; bhv[t] = bh[t]; }

  const int j     = tj * 16 + l16;
  const int ibase = ti * 16 + ((lane >> 4) << 3);
  constexpr size_t plane = (size_t)N * N;

#pragma unroll
  for (int r = 0; r < 8; ++r) {
    const float  s      = c[r] + bb;
    const size_t rowoff = (size_t)(ibase + r) * N + j;
#pragma unroll
    for (int t = 0; t < T; ++t) {
      out[(size_t)(b * T + t) * plane + rowoff] = fmaxf(fmaf(s, whv[t], bhv[t]), 0.f);
    }
  }
}

// ---------------------------------------------------------------------------
extern "C" void kernel_launch(void* const* d_in, const int* in_sizes, int n_in,
                              void* d_out, int out_size, void* d_ws, size_t ws_size,
                              hipStream_t stream) {
  const float* X   = (const float*)d_in[0];   // (4,1024,256) -> (4096,256)
  const float* W1o = (const float*)d_in[1];
  const float* b1o = (const float*)d_in[2];
  const float* W2o = (const float*)d_in[3];
  const float* b2o = (const float*)d_in[4];
  const float* W1d = (const float*)d_in[5];
  const float* b1d = (const float*)d_in[6];
  const float* W2d = (const float*)d_in[7];
  const float* b2d = (const float*)d_in[8];
  const float* Wb  = (const float*)d_in[9];
  const float* bb  = (const float*)d_in[10];
  const float* wh  = (const float*)d_in[11];
  const float* bh  = (const float*)d_in[12];
  float* out = (float*)d_out;

  constexpr int BN = 4096, H = 256, H2 = 128;

  // Workspace layout (fp32): ~12.7 MB total, resident in L2.
  float* ws    = (float*)d_ws;
  float* W2oWb = ws;                      // 256*128 : W2o @ Wb
  float* b2oWb = W2oWb + H * H2;          // 128     : b2o @ Wb
  float* hidO  = b2oWb + H2;              // 4096*256
  float* hidD  = hidO + (size_t)BN * H;   // 4096*256
  float* oWm   = hidD + (size_t)BN * H;   // 4096*128
  float* dMm   = oWm + (size_t)BN * H2;   // 4096*128

  // 1) Fold bilinear weight into the origin head: W2o' = W2o @ Wb, b2o' = b2o @ Wb
  {
    const int tiles = (H / 16) * (H2 / 16);  // 128 tiles
    gemm_wmma_kernel<<<(tiles + 7) / 8, 256, 0, stream>>>(W2o, Wb, nullptr, W2oWb,
                                                          H, H2, H2, 0);
  }
  vecmat128_kernel<<<1, 128, 0, stream>>>(b2o, Wb, b2oWb);

  // 2) Encoder hidden layers: relu(X @ W1 + b1), (4096 x 256 x 256)
  {
    const int tiles = (BN / 16) * (H / 16);  // 4096 tiles
    gemm_wmma_kernel<<<(tiles + 7) / 8, 256, 0, stream>>>(X, W1o, b1o, hidO,
                                                          BN, H, H, 1);
    gemm_wmma_kernel<<<(tiles + 7) / 8, 256, 0, stream>>>(X, W1d, b1d, hidD,
                                                          BN, H, H, 1);
  }

  // 3) Second layers: oW = hidden_o @ W2o' + b2o', d = hidden_d @ W2d + b2d
  {
    const int tiles = (BN / 16) * (H2 / 16);  // 2048 tiles
    gemm_wmma_kernel<<<(tiles + 7) / 8, 256, 0, stream>>>(hidO, W2oWb, b2oWb, oWm,
                                                          BN, H2, H, 0);
    gemm_wmma_kernel<<<(tiles + 7) / 8, 256, 0, stream>>>(hidD, W2d, b2d, dMm,
                                                          BN, H2, H, 0);
  }

  // 4) Fused bilinear scores + 12-horizon expansion -> (B,T,N,N), 201 MB stream
  {
    const int waves = 4 * 64 * 64;  // B * (N/16)^2 tiles, one per wave
    scores_expand_kernel<<<waves / 8, 256, 0, stream>>>(oWm, dMm, wh, bh, bb, out);
  }
}